// IoU3DLoss_23639499997213
// MI455X (gfx1250) — compile-verified
//
#include <hip/hip_runtime.h>
#include <hip/hip_bf16.h>

#define IOU_OFFSET 1e-06f
#define IOU_EPS    1e-06f

typedef __attribute__((ext_vector_type(2))) float v2f;
typedef __attribute__((ext_vector_type(8))) float v8f;

// Exact f32 full-wave (32-lane) sum using two V_WMMA_F32_16X16X4_F32 ops with
// B = all-ones. After WMMA#1, D[v] (uniform over columns) = val[v] + val[v+16]
// for lanes 0-15 and val[v+8] + val[v+24] for lanes 16-31. Summing the 8 D
// VGPRs gives a half-wave-uniform partial t; WMMA#2 then yields
// D[0] = t_lo + t_hi = full wave sum, broadcast to all 32 lanes.
// Requires EXEC == all ones (guaranteed by callers below).
__device__ __forceinline__ float wmma_wave_sum_f32(float val) {
    v2f a;  a.x = val;  a.y = 0.0f;
    v2f b;  b.x = 1.0f; b.y = 1.0f;   // all-ones B works for any B layout
    v8f c = {};
    c = __builtin_amdgcn_wmma_f32_16x16x4_f32(false, a, false, b,
                                              (short)0, c, false, false);
    float t = c[0] + c[1] + c[2] + c[3] + c[4] + c[5] + c[6] + c[7];
    v2f a2; a2.x = t; a2.y = 0.0f;
    v8f d = {};
    d = __builtin_amdgcn_wmma_f32_16x16x4_f32(false, a2, false, b,
                                              (short)0, d, false, false);
    return d[0];   // full 32-lane sum, present in every lane
}

// One thread per matched (pred[i], target[i]) pair. Computes the diagonal
// IoU3D term, then reduces loss-sum and valid-count per block via WMMA wave
// reductions + LDS, writing (loss, count) per block to `partials`.
__global__ __launch_bounds__(256)
void iou3d_diag_partial_kernel(const float* __restrict__ pred,
                               const float* __restrict__ tgt,
                               float* __restrict__ partials, int m) {
    const int i = blockIdx.x * blockDim.x + threadIdx.x;

    float loss = 0.0f;
    float cnt  = 0.0f;

    if (i < m) {
        const float* p = pred + (size_t)i * 7;
        const float* t = tgt  + (size_t)i * 7;
        const float px = p[0], py = p[1], pz = p[2];
        const float pw = p[3], pl = p[4], ph = p[5], pr = p[6];
        const float tx = t[0], ty = t[1], tz = t[2];
        const float tw = t[3], tl = t[4], th = t[5], tr = t[6];

        const bool valid = (pw > 0.0f) && (pl > 0.0f) && (ph > 0.0f);

        // Rotated BEV box -> axis-aligned standup half-extents (exact).
        const float cp = __cosf(pr), sp = __sinf(pr);
        const float ct = __cosf(tr), st = __sinf(tr);
        const float pex = 0.5f * (fabsf(pw * cp) + fabsf(pl * sp));
        const float pey = 0.5f * (fabsf(pw * sp) + fabsf(pl * cp));
        const float tex = 0.5f * (fabsf(tw * ct) + fabsf(tl * st));
        const float tey = 0.5f * (fabsf(tw * st) + fabsf(tl * ct));

        // BEV overlap of standup boxes (diagonal element only).
        const float ltx = fmaxf(px - pex, tx - tex);
        const float lty = fmaxf(py - pey, ty - tey);
        const float rbx = fminf(px + pex, tx + tex);
        const float rby = fminf(py + pey, ty + tey);
        const float wx = fmaxf(rbx - ltx + IOU_OFFSET, 0.0f);
        const float wy = fmaxf(rby - lty + IOU_OFFSET, 0.0f);
        const float ov_bev = wx * wy;

        // Height overlap (z is box top).
        const float ov_h = fmaxf(fminf(pz, tz) - fmaxf(pz - ph, tz - th), 0.0f);

        const float ov3   = ov_bev * ov_h;
        const float pvol  = pw * pl * ph;
        const float tvol  = tw * tl * th;
        const float denom = fmaxf(pvol + tvol - ov3, IOU_EPS);
        const float iou   = fminf(fmaxf(ov3 / denom, IOU_EPS), 1.0f);

        if (valid) {
            loss = -__logf(iou);
            cnt  = 1.0f;
        }
    }

    // --- reduction: wave (WMMA) -> block (LDS) ---
    __shared__ float s_loss[8];
    __shared__ float s_cnt[8];

    const int wave = threadIdx.x >> 5;
    const int lane = threadIdx.x & 31;

    const float wls = wmma_wave_sum_f32(loss);
    const float wcs = wmma_wave_sum_f32(cnt);

    if (lane == 0) {
        s_loss[wave] = wls;
        s_cnt[wave]  = wcs;
    }
    __syncthreads();

    if (threadIdx.x == 0) {
        const int nw = blockDim.x >> 5;
        float L = 0.0f, C = 0.0f;
        for (int w = 0; w < nw; ++w) { L += s_loss[w]; C += s_cnt[w]; }
        partials[2 * blockIdx.x + 0] = L;
        partials[2 * blockIdx.x + 1] = C;
    }
}

// Single-wave finalize: strided load of per-block partials, WMMA wave
// reduction, then the scalar epilogue matching the reference.
__global__ __launch_bounds__(32)
void iou3d_final_kernel(const float* __restrict__ partials, int nblocks,
                        float* __restrict__ out) {
    const int lane = threadIdx.x;   // 32 lanes, all active
    float L = 0.0f, C = 0.0f;
    for (int b = lane; b < nblocks; b += 32) {
        L += partials[2 * b + 0];
        C += partials[2 * b + 1];
    }
    const float tl = wmma_wave_sum_f32(L);
    const float tc = wmma_wave_sum_f32(C);
    if (lane == 0) {
        const float loss = (tc > 0.0f) ? (tl / fmaxf(tc, 1.0f)) : 0.0f;
        out[0] = loss * 1.0f;   // LOSS_WEIGHT
    }
}

extern "C" void kernel_launch(void* const* d_in, const int* in_sizes, int n_in,
                              void* d_out, int out_size, void* d_ws, size_t ws_size,
                              hipStream_t stream) {
    const float* pred = (const float*)d_in[0];
    const float* tgt  = (const float*)d_in[1];
    float* out = (float*)d_out;
    float* partials = (float*)d_ws;

    const int m = in_sizes[0] / 7;          // (m, 7) float32
    const int threads = 256;
    const int nblocks = (m + threads - 1) / threads;   // 16 for m = 4096

    iou3d_diag_partial_kernel<<<nblocks, threads, 0, stream>>>(pred, tgt, partials, m);
    iou3d_final_kernel<<<1, 32, 0, stream>>>(partials, nblocks, out);
}